// RGBBlock_47270410060035
// MI455X (gfx1250) — compile-verified
//
#include <hip/hip_runtime.h>

// ---------------------------------------------------------------------------
// RGBBlock (StyleGAN2-style to_rgb): style linear + per-sample modulated 1x1
// conv + residual add.
//   Stage 1: style = istyle @ style_w^T + style_b ; mod = style + 1
//            -> true 16x256x512 fp32 GEMM, done with v_wmma_f32_16x16x4_f32
//   Stage 2: out[b,o,hw] = sum_c conv_w[o,c]*mod[b,c]*x[b,c,hw] + prev_rgb
//            -> memory-bound streaming reduction (x = 256 MB read once),
//               float4 NT loads, weights broadcast from LDS.
// ---------------------------------------------------------------------------

typedef __attribute__((ext_vector_type(2))) float v2f;
typedef __attribute__((ext_vector_type(4))) float f4;
typedef __attribute__((ext_vector_type(8))) float v8f;

#define N_BATCH  16
#define N_CH     256
#define N_LAT    512
#define N_OUT    3
#define HW       (128 * 128)
#define HW4      (HW / 4)          // float4 elements per plane = 4096

// ---------------------------------------------------------------------------
// Stage 1: style GEMM via WMMA f32 16x16x4.
// One workgroup, 16 waves; wave w computes the 16(batch) x 16(channel) tile
// of channels [16w, 16w+16), accumulating over K = 512 in steps of 4.
//
// fp32 A 16x4 layout (ISA 7.12.2): lanes 0-15 / 16-31 both hold M=0..15;
//   VGPR0 = K=0 (lo lanes) / K=2 (hi lanes); VGPR1 = K=1 / K=3.
// => per lane, the two A elements are CONTIGUOUS in K -> 8-byte loads.
// B (4x16) mirrors this with lanes indexing column N.
// ---------------------------------------------------------------------------
__global__ __launch_bounds__(512) void rgb_style_wmma(
    const float* __restrict__ istyle,    // [16, 512]
    const float* __restrict__ style_w,   // [256, 512]
    const float* __restrict__ style_b,   // [256]
    float* __restrict__ mod)             // [16, 256] workspace (style + 1)
{
    const int lane = threadIdx.x & 31;
    const int wave = threadIdx.x >> 5;        // 0..15 -> channel tile
    const int n0   = wave * 16;
    const int m    = lane & 15;               // A row / B column within tile
    const int koff = (lane >> 4) << 1;        // 0 (lanes 0-15) or 2 (16-31)

    const float* __restrict__ arow = istyle  + (size_t)m * N_LAT;          // A row m
    const float* __restrict__ brow = style_w + (size_t)(n0 + m) * N_LAT;   // B col n0+m

    v8f acc = {};
#pragma unroll 4
    for (int k0 = 0; k0 < N_LAT; k0 += 4) {
        v2f a = *(const v2f*)(arow + k0 + koff);    // global_load_b64
        v2f b = *(const v2f*)(brow + k0 + koff);    // global_load_b64
        // (neg_a, A, neg_b, B, c_mod, C, reuse_a, reuse_b)
        acc = __builtin_amdgcn_wmma_f32_16x16x4_f32(
            false, a, false, b, (short)0, acc, false, false);
    }

    // D layout: VGPR j, lanes 0-15 -> row M=j, lanes 16-31 -> row M=j+8;
    // column N = lane & 15 (== m). Fuse bias and the "+1" modulation.
    const float bias  = style_b[n0 + m] + 1.0f;
    const int   rbase = (lane < 16) ? 0 : 8;
#pragma unroll
    for (int j = 0; j < 8; ++j) {
        mod[(size_t)(rbase + j) * N_CH + n0 + m] = acc[j] + bias;
    }
}

// ---------------------------------------------------------------------------
// Stage 2: streaming modulated 1x1 conv + residual.
// grid = 16 batches * 16 tiles, block = 256 threads (8 waves).
// Each thread owns one float4 (4 pixels); inner loop over 256 channels:
//   1 NT global_load_b128 (x) + 1 ds_load_b128 (weights) + 12 FMAs.
// x is streamed exactly once: 256 MB -> the ~11 us HBM roofline.
// ---------------------------------------------------------------------------
__global__ __launch_bounds__(256) void rgb_conv_stream(
    const float* __restrict__ x,        // [16, 256, 16384]
    const float* __restrict__ prev,     // [16, 3, 16384]
    const float* __restrict__ conv_w,   // [3, 256]
    const float* __restrict__ mod,      // [16, 256]
    float* __restrict__ out)            // [16, 3, 16384]
{
    __shared__ f4 wlds[N_CH];           // {w_r, w_g, w_b, 0} per channel: 4 KB

    const int b    = blockIdx.x >> 4;   // batch
    const int tile = blockIdx.x & 15;   // 1024-pixel tile within the plane
    const int tid  = threadIdx.x;

    // Build per-sample effective weights once (uniform across the block).
    {
        const float mv = mod[(size_t)b * N_CH + tid];
        f4 w;
        w.x = conv_w[0 * N_CH + tid] * mv;
        w.y = conv_w[1 * N_CH + tid] * mv;
        w.z = conv_w[2 * N_CH + tid] * mv;
        w.w = 0.0f;
        wlds[tid] = w;
    }
    __syncthreads();

    const int p = tile * 256 + tid;     // float4 index within a 4096-wide plane
    const f4* __restrict__ xp =
        (const f4*)x + (size_t)b * N_CH * HW4 + p;

    f4 a0 = {}, a1 = {}, a2 = {};
#pragma unroll 8
    for (int c = 0; c < N_CH; ++c) {
        f4 v = __builtin_nontemporal_load(xp + (size_t)c * HW4);
        f4 w = wlds[c];
        a0 += w.x * v;
        a1 += w.y * v;
        a2 += w.z * v;
    }

    const f4* __restrict__ pp = (const f4*)prev + (size_t)(b * N_OUT) * HW4 + p;
    f4* __restrict__       op = (f4*)out        + (size_t)(b * N_OUT) * HW4 + p;

    __builtin_nontemporal_store(a0 + pp[0 * HW4], op + 0 * HW4);
    __builtin_nontemporal_store(a1 + pp[1 * HW4], op + 1 * HW4);
    __builtin_nontemporal_store(a2 + pp[2 * HW4], op + 2 * HW4);
}

// ---------------------------------------------------------------------------
extern "C" void kernel_launch(void* const* d_in, const int* in_sizes, int n_in,
                              void* d_out, int out_size, void* d_ws, size_t ws_size,
                              hipStream_t stream) {
    const float* x       = (const float*)d_in[0];   // [16,256,128,128]
    const float* prev    = (const float*)d_in[1];   // [16,3,128,128]
    const float* istyle  = (const float*)d_in[2];   // [16,512]
    const float* style_w = (const float*)d_in[3];   // [256,512]
    const float* style_b = (const float*)d_in[4];   // [256]
    const float* conv_w  = (const float*)d_in[5];   // [3,256]
    float*       out     = (float*)d_out;
    float*       mod     = (float*)d_ws;            // [16,256] = 16 KB scratch

    // Stage 1: tiny WMMA GEMM (1 workgroup, 16 waves).
    rgb_style_wmma<<<1, 512, 0, stream>>>(istyle, style_w, style_b, mod);

    // Stage 2: bandwidth-bound streaming pass (256 blocks x 256 threads).
    rgb_conv_stream<<<N_BATCH * 16, 256, 0, stream>>>(x, prev, conv_w, mod, out);
}